// SparseLinear_48155173323398
// MI455X (gfx1250) — compile-verified
//
#include <hip/hip_runtime.h>

typedef __attribute__((ext_vector_type(2))) float v2f;
typedef __attribute__((ext_vector_type(4))) float v4f;
typedef __attribute__((ext_vector_type(8))) float v8f;

// ---------------------------------------------------------------------------
// WMMA 16x16 f32 lane-transpose building block (gfx1250):
//   A-operand layout: lane = matrix ROW   (lane m holds A[m][k])
//   D-operand layout: lane = matrix COL   (lane n+16h holds D[8h+r][n] in vgpr r)
// D = T x I (4 chained V_WMMA_F32_16X16X4_F32 with identity slices in B) is
// mathematically T but lands transposed across lanes. Exact in f32.
// ---------------------------------------------------------------------------

// Transpose x (B x N row-major) -> xT (N x B row-major), one 16x16 tile / wave.
__global__ void k_transpose_x(const float* __restrict__ x, float* __restrict__ xT,
                              int B, int N, int ntiles) {
  int wave = (int)((blockIdx.x * blockDim.x + threadIdx.x) >> 5);
  int lane = threadIdx.x & 31;
  if (wave >= ntiles) return;              // wave-uniform: EXEC stays all-ones
  int btiles = B >> 4;
  int b0 = (wave % btiles) << 4;
  int n0 = (wave / btiles) << 4;
  int m = lane & 15;                       // A row index for this lane
  int h = lane >> 4;                       // lane half selects K pair
  const float* src = x + (size_t)(b0 + m) * N + n0;
  v8f acc = {};
#pragma unroll
  for (int c = 0; c < 4; ++c) {
    int col = 4 * c + 2 * h;               // global K of this lane's vgpr0
    v2f a;  a[0] = src[col];  a[1] = src[col + 1];
    v2f bi; bi[0] = (m == col)     ? 1.0f : 0.0f;   // identity slice B[k][n]=d(n==k)
            bi[1] = (m == col + 1) ? 1.0f : 0.0f;
    acc = __builtin_amdgcn_wmma_f32_16x16x4_f32(false, a, false, bi,
                                                (short)0, acc, false, false);
  }
  // lane holds column n=m of the tile: D[8h+r][m] = x[b0+8h+r][n0+m]
  float* dst = xT + (size_t)(n0 + m) * B + b0 + 8 * h;
  *(v4f*)(dst)     = v4f{acc[0], acc[1], acc[2], acc[3]};
  *(v4f*)(dst + 4) = v4f{acc[4], acc[5], acc[6], acc[7]};
}

// Transpose outT (M x B row-major) -> out (B x M row-major), same trick.
__global__ void k_transpose_out(const float* __restrict__ outT, float* __restrict__ out,
                                int B, int M, int ntiles) {
  int wave = (int)((blockIdx.x * blockDim.x + threadIdx.x) >> 5);
  int lane = threadIdx.x & 31;
  if (wave >= ntiles) return;
  int btiles = B >> 4;
  int b0 = (wave % btiles) << 4;
  int m0 = (wave / btiles) << 4;
  int r = lane & 15;
  int h = lane >> 4;
  const float* src = outT + (size_t)(m0 + r) * B + b0;
  v8f acc = {};
#pragma unroll
  for (int c = 0; c < 4; ++c) {
    int col = 4 * c + 2 * h;
    v2f a;  a[0] = src[col];  a[1] = src[col + 1];
    v2f bi; bi[0] = (r == col)     ? 1.0f : 0.0f;
            bi[1] = (r == col + 1) ? 1.0f : 0.0f;
    acc = __builtin_amdgcn_wmma_f32_16x16x4_f32(false, a, false, bi,
                                                (short)0, acc, false, false);
  }
  // lane holds D[8h+k][r] = outT[m0+8h+k][b0+r]  ->  out[b0+r][m0+8h+k]
  float* dst = out + (size_t)(b0 + r) * M + m0 + 8 * h;
  *(v4f*)(dst)     = v4f{acc[0], acc[1], acc[2], acc[3]};
  *(v4f*)(dst + 4) = v4f{acc[4], acc[5], acc[6], acc[7]};
}

// outT[m][0..B) = bias[m]  (float4-vectorized broadcast init)
__global__ void k_init_outT(const float* __restrict__ bias, float* __restrict__ outT,
                            int B, int M) {
  int tid = blockIdx.x * blockDim.x + threadIdx.x;
  int total = (M * B) >> 2;
  if (tid >= total) return;
  int per_row = B >> 2;                    // float4s per m-row (8 for B=32)
  float bv = bias[tid / per_row];
  ((v4f*)outT)[tid] = v4f{bv, bv, bv, bv};
}

// Hot kernel: each wave loads 32 edges coalesced, then per edge does one
// fully-coalesced 128B gather from xT and one 128B line of packed f32
// atomics into outT (lane == batch, B == 32 == wave32 width).
//
// Two-phase structure: phase 1 issues all 32 independent gathers back-to-back
// (one loadcnt wait total instead of 32 dependent load->wait->atomic chains);
// phase 2 fires 32 no-return f32 atomics (STOREcnt only, no in-loop waits).
// Edge broadcast uses v_readlane with compile-time lane index -> SGPR, so
// addresses are scalar-base + lane offset (SADDR VMEM, SALU address math).
__global__ void k_edge_scatter(const long long* __restrict__ srcI,
                               const long long* __restrict__ dstI,
                               const float* __restrict__ vals,
                               const float* __restrict__ xT,
                               float* __restrict__ outT,
                               int B, int E) {
  int e = blockIdx.x * blockDim.x + threadIdx.x;
  int lane = threadIdx.x & 31;
  int s = 0, d = 0;
  float v = 0.0f;
  if (e < E) {                             // E % 256 == 0 here, but stay safe
    s = (int)srcI[e];
    d = (int)dstI[e];
    v = vals[e];
  }
  int vbits = __float_as_int(v);

  float xv[32];
#pragma unroll
  for (int i = 0; i < 32; ++i) {
    int si = __builtin_amdgcn_readlane(s, i);          // wave-uniform (SGPR)
    xv[i] = xT[(size_t)si * B + lane];                 // 32 independent gathers
  }
#pragma unroll
  for (int i = 0; i < 32; ++i) {
    int   di = __builtin_amdgcn_readlane(d, i);
    float vi = __int_as_float(__builtin_amdgcn_readlane(vbits, i));
    atomicAdd(&outT[(size_t)di * B + lane], vi * xv[i]);  // packed line atomics
  }
}

extern "C" void kernel_launch(void* const* d_in, const int* in_sizes, int n_in,
                              void* d_out, int out_size, void* d_ws, size_t ws_size,
                              hipStream_t stream) {
  const float*     x    = (const float*)d_in[0];
  const long long* idx  = (const long long*)d_in[1];   // int64 indices (2, E)
  const float*     vals = (const float*)d_in[2];
  const float*     bias = (const float*)d_in[3];
  float*           out  = (float*)d_out;

  const int E = in_sizes[2];
  const int M = in_sizes[3];
  const int B = out_size / M;          // 32
  const int N = in_sizes[0] / B;       // 50000

  float* xT   = (float*)d_ws;                  // N*B floats (6.4 MB)
  float* outT = xT + (size_t)N * B;            // M*B floats (6.4 MB)

  const long long* srcI = idx;
  const long long* dstI = idx + E;

  const int threads = 256;
  const int wpb = threads / 32;

  // 1) x -> xT (WMMA lane-transpose, one wave per 16x16 tile)
  int xtiles = (N / 16) * (B / 16);
  k_transpose_x<<<dim3((xtiles + wpb - 1) / wpb), dim3(threads), 0, stream>>>(
      x, xT, B, N, xtiles);

  // 2) outT[m][:] = bias[m]
  int total4 = (M * B) / 4;
  k_init_outT<<<dim3((total4 + threads - 1) / threads), dim3(threads), 0, stream>>>(
      bias, outT, B, M);

  // 3) edge scatter (dominant kernel): E/32 waves, one edge chunk each
  k_edge_scatter<<<dim3((E + threads - 1) / threads), dim3(threads), 0, stream>>>(
      srcI, dstI, vals, xT, outT, B, E);

  // 4) outT -> out (WMMA lane-transpose)
  int otiles = (M / 16) * (B / 16);
  k_transpose_out<<<dim3((otiles + wpb - 1) / wpb), dim3(threads), 0, stream>>>(
      outT, out, B, M, otiles);
}